// G4GCN_66838281060723
// MI455X (gfx1250) — compile-verified
//
#include <hip/hip_runtime.h>

// ---------------------------------------------------------------------------
// G4GCN on MI455X (gfx1250).
// Memory/atomic-bound graph conv: edge scatter via L2-resident f32 atomics,
// node update (6->2 linear) via V_WMMA_F32_16X16X4_F32 (exact f32).
// ---------------------------------------------------------------------------

typedef __attribute__((ext_vector_type(2))) float v2f;
typedef __attribute__((ext_vector_type(8))) float v8f;

__device__ __forceinline__ void atomAddF(float* p, float v) {
    // native global_atomic_add_f32, no return, agent scope (resolves in L2)
    __hip_atomic_fetch_add(p, v, __ATOMIC_RELAXED, __HIP_MEMORY_SCOPE_AGENT);
}

__device__ __forceinline__ v8f wmma_f32_16x16x4(v2f a, v2f b, v8f c) {
    // 8-arg pattern: (neg_a, A, neg_b, B, c_mod, C, reuse_a, reuse_b)
    return __builtin_amdgcn_wmma_f32_16x16x4_f32(
        false, a, false, b, (short)0, c, false, false);
}

// inv_deg = deg^-0.5 (0 when deg==0); x0 = x*W0 + b0  ([N,1] @ [1,2])
__global__ void init_nodes_kernel(const float* __restrict__ xin,
                                  const float* __restrict__ deg,
                                  const float* __restrict__ W0,
                                  const float* __restrict__ b0,
                                  float* __restrict__ inv,
                                  float* __restrict__ xout, int N) {
    int i = blockIdx.x * blockDim.x + threadIdx.x;
    if (i >= N) return;
    float d = deg[i];
    inv[i] = (d > 0.f) ? (1.0f / sqrtf(d)) : 0.f;
    float xi = xin[i];
    xout[2 * i + 0] = fmaf(xi, W0[0], b0[0]);
    xout[2 * i + 1] = fmaf(xi, W0[1], b0[1]);
}

// per-node incident-edge counts (layer-invariant self-term multiplier)
__global__ void count_kernel(const int* __restrict__ src,
                             const int* __restrict__ trg,
                             float* __restrict__ cnt_c,
                             float* __restrict__ cnt_v, int E) {
    int e = blockIdx.x * blockDim.x + threadIdx.x;
    if (e >= E) return;
    atomAddF(&cnt_c[src[e]], 1.0f);
    atomAddF(&cnt_v[trg[e]], 1.0f);
}

// Both directions of one edge list: norm is symmetric, so one pass scatters
// norm*xc[s] into v_acc[t] and norm*xv[t] into c_acc[s].
__global__ void edge_kernel(const int* __restrict__ src,
                            const int* __restrict__ trg,
                            const float* __restrict__ inv_c,
                            const float* __restrict__ inv_v,
                            const float* __restrict__ xc,
                            const float* __restrict__ xv,
                            float* __restrict__ c_acc,
                            float* __restrict__ v_acc, int E) {
    int e = blockIdx.x * blockDim.x + threadIdx.x;
    if (e >= E) return;
    int s = src[e];
    int t = trg[e];
    float norm = inv_c[s] * inv_v[t];
    float xc0 = xc[2 * s], xc1 = xc[2 * s + 1];
    float xv0 = xv[2 * t], xv1 = xv[2 * t + 1];
    atomAddF(&v_acc[2 * t + 0], norm * xc0);
    atomAddF(&v_acc[2 * t + 1], norm * xc1);
    atomAddF(&c_acc[2 * s + 0], norm * xv0);
    atomAddF(&c_acc[2 * s + 1], norm * xv1);
}

// Node update: X = [pos_acc + cnt_pos*x, neg_acc + cnt_neg*x, x]  ([16,6])
// Y = X @ W + b via two V_WMMA_F32_16X16X4_F32 (K = 0..3, then 4..5 + zeros).
// One wave per 16 nodes. Lanes 0-15 carry K=0,1; lanes 16-31 carry K=2,3
// (A 16x4 f32 layout); C initialized to bias; only D columns 0,1 stored.
__global__ void update_wmma_kernel(const float* __restrict__ x,
                                   const float* __restrict__ pos_acc,
                                   const float* __restrict__ neg_acc,
                                   const float* __restrict__ cnt_pos,
                                   const float* __restrict__ cnt_neg,
                                   const float* __restrict__ W,   // [6,2]
                                   const float* __restrict__ b,   // [2]
                                   float* __restrict__ out, int N) {
    int lane = threadIdx.x & 31;
    int wave = threadIdx.x >> 5;
    int base = (blockIdx.x * (blockDim.x >> 5) + wave) << 4;  // 16 nodes/wave
    int m = lane & 15;
    bool hi = lane >= 16;
    int node = base + m;
    int nc = (node < N) ? node : 0;  // clamp: safe loads for pad rows

    float x0 = x[2 * nc + 0];
    float x1 = x[2 * nc + 1];

    // Branchless pointer selects (all addresses valid for every lane).
    const float* cntp = hi ? cnt_neg : cnt_pos;
    const float* accp = hi ? neg_acc : pos_acc;
    float cnt = cntp[nc];

    // A for WMMA #1: K=0,1 (pos agg) in lanes 0-15; K=2,3 (neg agg) in 16-31
    v2f A1;
    A1.x = fmaf(cnt, x0, accp[2 * nc + 0]);
    A1.y = fmaf(cnt, x1, accp[2 * nc + 1]);
    // A for WMMA #2: K=4,5 (prev state) in lanes 0-15; K=6,7 are zero padding
    v2f A2;
    A2.x = hi ? 0.f : x0;
    A2.y = hi ? 0.f : x1;

    // B: column n = lane&15 (only n<2 real). Lanes 0-15: K=0,1 / 16-31: K=2,3
    // Branchless: clamp column index, mask value (W has 12 elems, idx<=11).
    int n = lane & 15;
    bool live = n < 2;
    int ncl = live ? n : 0;
    float mask = live ? 1.f : 0.f;
    int kh = hi ? 2 : 0;
    v2f B1;
    B1.x = W[(kh + 0) * 2 + ncl] * mask;
    B1.y = W[(kh + 1) * 2 + ncl] * mask;
    float m2 = (live && !hi) ? 1.f : 0.f;
    v2f B2;
    B2.x = W[4 * 2 + ncl] * m2;
    B2.y = W[5 * 2 + ncl] * m2;
    float cb = b[ncl] * mask;

    v8f C;
#pragma unroll
    for (int r = 0; r < 8; ++r) C[r] = cb;  // bias as initial accumulator

    C = wmma_f32_16x16x4(A1, B1, C);  // K = 0..3
    C = wmma_f32_16x16x4(A2, B2, C);  // K = 4..5 (+ zero K=6,7)

    // D layout: VGPR r, lanes 0-15 -> (M=r, N=lane); lanes 16-31 -> (M=r+8).
    // `base` is wave-uniform: the tile-in-range test is a scalar branch, so
    // the common path is 8 unpredicated stores (no per-element EXEC churn).
    int mb = base + (hi ? 8 : 0);
    if (base + 16 <= N) {
        if (live) {
#pragma unroll
            for (int r = 0; r < 8; ++r) out[2 * (mb + r) + n] = C[r];
        }
    } else {
        if (live) {
#pragma unroll
            for (int r = 0; r < 8; ++r) {
                int mm = mb + r;
                if (mm < N) out[2 * mm + n] = C[r];
            }
        }
    }
}

extern "C" void kernel_launch(void* const* d_in, const int* in_sizes, int n_in,
                              void* d_out, int out_size, void* d_ws, size_t ws_size,
                              hipStream_t stream) {
    const float* x_clause   = (const float*)d_in[0];
    const float* x_variable = (const float*)d_in[1];
    const float* deg_clause   = (const float*)d_in[2];
    const float* deg_variable = (const float*)d_in[3];
    const int* pos_src = (const int*)d_in[4];
    const int* pos_trg = (const int*)d_in[5];
    const int* neg_src = (const int*)d_in[6];
    const int* neg_trg = (const int*)d_in[7];
    const float* W0c = (const float*)d_in[8];
    const float* b0c = (const float*)d_in[9];
    const float* W0v = (const float*)d_in[10];
    const float* b0v = (const float*)d_in[11];
    const float* Wc  = (const float*)d_in[12];  // [4,6,2]
    const float* bc  = (const float*)d_in[13];  // [4,2]
    const float* Wv  = (const float*)d_in[14];  // [4,6,2]
    const float* bv  = (const float*)d_in[15];  // [4,2]

    const int NCn = in_sizes[0];
    const int NVn = in_sizes[1];
    const int En  = in_sizes[4];

    // ---- workspace carve-up (~110 MB of f32) ----
    float* ws = (float*)d_ws;
    size_t o = 0;
    float* inv_c = ws + o;  o += (size_t)NCn;
    float* inv_v = ws + o;  o += (size_t)NVn;
    float* cnt_blk   = ws + o;
    float* cnt_c_pos = ws + o;  o += (size_t)NCn;
    float* cnt_c_neg = ws + o;  o += (size_t)NCn;
    float* cnt_v_pos = ws + o;  o += (size_t)NVn;
    float* cnt_v_neg = ws + o;  o += (size_t)NVn;
    const size_t cnt_floats = 2 * (size_t)NCn + 2 * (size_t)NVn;
    float* xcA = ws + o;  o += 2 * (size_t)NCn;
    float* xcB = ws + o;  o += 2 * (size_t)NCn;
    float* xvA = ws + o;  o += 2 * (size_t)NVn;
    float* xvB = ws + o;  o += 2 * (size_t)NVn;
    float* acc_blk   = ws + o;
    float* v_pos_acc = ws + o;  o += 2 * (size_t)NVn;
    float* v_neg_acc = ws + o;  o += 2 * (size_t)NVn;
    float* c_pos_acc = ws + o;  o += 2 * (size_t)NCn;
    float* c_neg_acc = ws + o;  o += 2 * (size_t)NCn;
    const size_t acc_floats = 4 * (size_t)NVn + 4 * (size_t)NCn;
    (void)ws_size; (void)n_in;

    const int BT = 256;  // 8 wave32

    // ---- one-time (per launch) precompute ----
    hipMemsetAsync(cnt_blk, 0, cnt_floats * sizeof(float), stream);
    init_nodes_kernel<<<(NCn + BT - 1) / BT, BT, 0, stream>>>(
        x_clause, deg_clause, W0c, b0c, inv_c, xcA, NCn);
    init_nodes_kernel<<<(NVn + BT - 1) / BT, BT, 0, stream>>>(
        x_variable, deg_variable, W0v, b0v, inv_v, xvA, NVn);
    count_kernel<<<(En + BT - 1) / BT, BT, 0, stream>>>(
        pos_src, pos_trg, cnt_c_pos, cnt_v_pos, En);
    count_kernel<<<(En + BT - 1) / BT, BT, 0, stream>>>(
        neg_src, neg_trg, cnt_c_neg, cnt_v_neg, En);

    // ---- 4 GNN layers ----
    float* xc_cur = xcA; float* xc_nxt = xcB;
    float* xv_cur = xvA; float* xv_nxt = xvB;
    const int nodesPerBlock = (BT / 32) * 16;  // 128
    for (int l = 0; l < 4; ++l) {
        hipMemsetAsync(acc_blk, 0, acc_floats * sizeof(float), stream);
        edge_kernel<<<(En + BT - 1) / BT, BT, 0, stream>>>(
            pos_src, pos_trg, inv_c, inv_v, xc_cur, xv_cur,
            c_pos_acc, v_pos_acc, En);
        edge_kernel<<<(En + BT - 1) / BT, BT, 0, stream>>>(
            neg_src, neg_trg, inv_c, inv_v, xc_cur, xv_cur,
            c_neg_acc, v_neg_acc, En);

        float* v_out = (l == 3) ? (float*)d_out : xv_nxt;
        update_wmma_kernel<<<(NVn + nodesPerBlock - 1) / nodesPerBlock, BT, 0, stream>>>(
            xv_cur, v_pos_acc, v_neg_acc, cnt_v_pos, cnt_v_neg,
            Wv + (size_t)l * 12, bv + (size_t)l * 2, v_out, NVn);
        if (l < 3) {  // final clause state is never consumed
            update_wmma_kernel<<<(NCn + nodesPerBlock - 1) / nodesPerBlock, BT, 0, stream>>>(
                xc_cur, c_pos_acc, c_neg_acc, cnt_c_pos, cnt_c_neg,
                Wc + (size_t)l * 12, bc + (size_t)l * 2, xc_nxt, NCn);
        }
        float* t;
        t = xc_cur; xc_cur = xc_nxt; xc_nxt = t;
        t = xv_cur; xv_cur = xv_nxt; xv_nxt = t;
    }
}